// HDCMemoryEMA_58334245814672
// MI455X (gfx1250) — compile-verified
//
#include <hip/hip_runtime.h>
#include <hip/hip_bf16.h>

// EMA scan  c[b,t,d] = x[b,t,d] + decay * c[b,t-1,d]
// B=8, T=4096, D=2048, fp32.
//
// Chunked-scan formulation: within a 16-step chunk,
//   y_intra = L @ X      (L[t][s] = decay^(t-s), lower triangular 16x16)
// computed with 4x V_WMMA_F32_16X16X4_F32, then
//   y[t] = y_intra[t] + decay^(t+1) * carry_in ;  carry_out = y[15].
// T is split into NS=16 segments of S=256 for parallelism; segment carries
// are computed in a cheap first pass and prefix-combined in a tiny pass.

#define B_  8
#define T_  4096
#define D_  2048
#define S_  256          // timesteps per segment
#define NS_ 16           // T_/S_
#define CH_ 16           // WMMA chunk (timesteps)
#define NCH_ 16          // S_/CH_

typedef float v2f __attribute__((ext_vector_type(2)));
typedef float v8f __attribute__((ext_vector_type(8)));

__device__ __forceinline__ float sigmoidf_(float z) {
    return 1.0f / (1.0f + __expf(-z));
}

// ---------------------------------------------------------------------------
// Pass 1: per-segment local carry-out  C[b][g][d] = sum_{t in seg} decay^(last-t) x[t]
// One wave = 32 consecutive d columns; loop over S_ timesteps (coalesced 128B).
// Waves: B_ * NS_ * (D_/32) = 8192.
// ---------------------------------------------------------------------------
__global__ void ema_seg_carry(const float* __restrict__ x,
                              const float* __restrict__ logit,
                              float* __restrict__ C) {
    const float decay = sigmoidf_(logit[0]);
    const int lane = threadIdx.x & 31;
    const int wid  = (blockIdx.x * blockDim.x + threadIdx.x) >> 5;
    const int DW = D_ / 32;                 // 64 wave-tiles across D
    const int dw = wid % DW;
    const int g  = (wid / DW) % NS_;
    const int b  = wid / (DW * NS_);
    const int d  = dw * 32 + lane;

    const float* xp = x + ((size_t)b * T_ + (size_t)g * S_) * D_ + d;
    float c = 0.0f;
#pragma unroll 8
    for (int t = 0; t < S_; ++t) {
        c = fmaf(decay, c, xp[(size_t)t * D_]);   // c = x_t + decay*c
    }
    C[((size_t)b * NS_ + g) * D_ + d] = c;
}

// ---------------------------------------------------------------------------
// Pass 2: exclusive prefix across segments per (b,d):
//   P[g] = state_{g-1},  state_g = decay^S * state_{g-1} + C[g],  state_{-1}=0
// ---------------------------------------------------------------------------
__global__ void ema_seg_prefix(const float* __restrict__ logit,
                               const float* __restrict__ C,
                               float* __restrict__ P) {
    const int idx = blockIdx.x * blockDim.x + threadIdx.x;   // over B_*D_
    if (idx >= B_ * D_) return;
    const int b = idx / D_;
    const int d = idx - b * D_;
    const float decay = sigmoidf_(logit[0]);
    const float dS = __powf(decay, (float)S_);
    float state = 0.0f;
    for (int g = 0; g < NS_; ++g) {
        const size_t o = ((size_t)b * NS_ + g) * D_ + d;
        P[o] = state;
        state = fmaf(dS, state, C[o]);
    }
}

// ---------------------------------------------------------------------------
// Pass 3: main WMMA chunked scan.
// One wave owns (b, 16-wide d tile, segment g): 16 chunks of 16 timesteps.
// Per chunk: 4x v_wmma_f32_16x16x4_f32 (L @ Xchunk), carry inject, store,
// carry update via one __shfl.
// Waves: B_ * (D_/16) * NS_ = 16384.
// Fragment layouts (ISA 7.12.2, f32):
//   A 16x4 (2 VGPR): m = lane&15 ; v0:K=2h, v1:K=2h+1  (h = lane>>4)
//   B 4x16 (2 VGPR): n = lane&15 ; v0:K=2h, v1:K=2h+1  (assumed mirror of A)
//   C 16x16 (8 VGPR): VGPR k -> row (k + 8h), col n = lane&15
// ---------------------------------------------------------------------------
__global__ void ema_wmma_scan(const float* __restrict__ x,
                              const float* __restrict__ logit,
                              const float* __restrict__ P,
                              float* __restrict__ out) {
    const float decay = sigmoidf_(logit[0]);
    const int lane = threadIdx.x & 31;
    const int wid  = (blockIdx.x * blockDim.x + threadIdx.x) >> 5;
    const int DT = D_ / 16;                  // 128 d-tiles
    const int dtile = wid % DT;
    const int g     = (wid / DT) % NS_;
    const int b     = wid / (DT * NS_);

    const int n = lane & 15;                 // column within tile (also m for A)
    const int h = lane >> 4;                 // half-wave
    const int d = dtile * 16 + n;

    // A fragments: L columns for the 4 K-steps. L[m][k] = decay^(m-k), k<=m.
    v2f Af[4];
#pragma unroll
    for (int j = 0; j < 4; ++j) {
        const int k0 = 4 * j + 2 * h;
        const int k1 = k0 + 1;
        Af[j].x = (k0 <= n) ? __powf(decay, (float)(n - k0)) : 0.0f;
        Af[j].y = (k1 <= n) ? __powf(decay, (float)(n - k1)) : 0.0f;
    }
    // Row scales for carry injection: rs[k] = decay^(row+1), row = k + 8h.
    float rs[8];
#pragma unroll
    for (int k = 0; k < 8; ++k)
        rs[k] = __powf(decay, (float)(k + 8 * h + 1));
    const float d16 = 0.0f; (void)d16;

    // Carry-in for this segment (replicated across both half-waves).
    float carry = P[((size_t)b * NS_ + g) * D_ + d];

    const size_t segRow = (size_t)b * T_ + (size_t)g * S_;

    for (int ch = 0; ch < NCH_; ++ch) {
        const float* xb = x + (segRow + (size_t)ch * CH_) * D_;
        float*       ob = out + (segRow + (size_t)ch * CH_) * D_;

        if (ch + 1 < NCH_) {
            // Pull next chunk toward the caches (global_prefetch_b8).
            __builtin_prefetch(xb + (size_t)(CH_ + 2 * h) * D_ + d, 0, 3);
            __builtin_prefetch(xb + (size_t)(CH_ + 8 + 2 * h) * D_ + d, 0, 3);
        }

        v8f acc = {0.0f, 0.0f, 0.0f, 0.0f, 0.0f, 0.0f, 0.0f, 0.0f};
#pragma unroll
        for (int j = 0; j < 4; ++j) {
            const int r0 = 4 * j + 2 * h;
            v2f Bf;
            Bf.x = xb[(size_t)r0 * D_ + d];
            Bf.y = xb[(size_t)(r0 + 1) * D_ + d];
            acc = __builtin_amdgcn_wmma_f32_16x16x4_f32(
                false, Af[j], false, Bf, (short)0, acc, false, false);
        }

        // Inject carry, store, grab row 15 for the next chunk's carry.
        float y7 = 0.0f;
#pragma unroll
        for (int k = 0; k < 8; ++k) {
            const float y = acc[k] + rs[k] * carry;
            ob[(size_t)(k + 8 * h) * D_ + d] = y;
            if (k == 7) y7 = y;
        }
        // Row 15 lives in VGPR7 of half-wave h=1 (lanes 16..31), col = lane&15.
        carry = __shfl(y7, lane | 16, 32);
    }
}

// ---------------------------------------------------------------------------
extern "C" void kernel_launch(void* const* d_in, const int* in_sizes, int n_in,
                              void* d_out, int out_size, void* d_ws, size_t ws_size,
                              hipStream_t stream) {
    const float* x     = (const float*)d_in[0];
    const float* logit = (const float*)d_in[1];
    float*       out   = (float*)d_out;

    // Workspace: C carries then P prefixes, each [B][NS][D] fp32 (1 MB each).
    float* C = (float*)d_ws;
    float* P = C + (size_t)B_ * NS_ * D_;

    // Pass 1: 8192 waves -> 262144 threads.
    {
        const int threads = B_ * NS_ * (D_ / 32) * 32;
        ema_seg_carry<<<threads / 256, 256, 0, stream>>>(x, logit, C);
    }
    // Pass 2: 16384 threads.
    {
        const int threads = B_ * D_;
        ema_seg_prefix<<<threads / 256, 256, 0, stream>>>(logit, C, P);
    }
    // Pass 3: 16384 waves -> 524288 threads.
    {
        const int threads = B_ * NS_ * (D_ / 16) * 32;
        ema_wmma_scan<<<threads / 256, 256, 0, stream>>>(x, logit, P, out);
    }
}